// VariableSelectionNetwork_84997402788316
// MI455X (gfx1250) — compile-verified
//
#include <hip/hip_runtime.h>
#include <hip/hip_bf16.h>

// Problem constants
#define BB 32
#define SS 512
#define FF 16
#define HH 256
#define BS (BB * SS)   // 16384 rows

typedef float v2f __attribute__((ext_vector_type(2)));
typedef float v8f __attribute__((ext_vector_type(8)));

__device__ __forceinline__ float elu_f(float v) {
    return v > 0.0f ? v : (__expf(v) - 1.0f);
}

// ---------------------------------------------------------------------------
// Phase 1: attention weights.  weights = softmax(elu(x@Wa1+ba1)@Wa2+ba2)
// One thread per row. Wa1 (transposed), Wa2, ba1, ba2 staged in LDS.
// Writes the [BS,16] weights block (second output) directly into d_out.
// ---------------------------------------------------------------------------
__global__ __launch_bounds__(256) void vsn_attn_kernel(
    const float* __restrict__ x, const float* __restrict__ Wa1,
    const float* __restrict__ ba1, const float* __restrict__ Wa2,
    const float* __restrict__ ba2, float* __restrict__ wout)
{
    __shared__ float s_wa1t[HH * FF];  // [h][f]
    __shared__ float s_wa2[HH * FF];   // [h][f] (native layout of Wa2)
    __shared__ float s_ba1[HH];
    __shared__ float s_ba2[FF];

    const int tid = threadIdx.x;
#pragma unroll
    for (int i = 0; i < 16; ++i) {
        int lin = tid + i * 256;          // over F*H = 4096
        int f = lin >> 8, h = lin & 255;
        s_wa1t[h * FF + f] = Wa1[lin];    // transpose Wa1[f][h] -> [h][f]
        s_wa2[lin] = Wa2[lin];            // Wa2 already [h][f]
    }
    s_ba1[tid] = ba1[tid & 255];
    if (tid < FF) s_ba2[tid] = ba2[tid];
    __syncthreads();

    const size_t row = (size_t)blockIdx.x * 256 + tid;
    const float4* xs = (const float4*)(x + row * FF);
    float4 x0 = xs[0], x1 = xs[1], x2 = xs[2], x3 = xs[3];
    float xv[16] = {x0.x, x0.y, x0.z, x0.w, x1.x, x1.y, x1.z, x1.w,
                    x2.x, x2.y, x2.z, x2.w, x3.x, x3.y, x3.z, x3.w};

    float lg[16];
#pragma unroll
    for (int f = 0; f < 16; ++f) lg[f] = s_ba2[f];

#pragma unroll 2
    for (int h = 0; h < HH; ++h) {
        const float4* wc = (const float4*)&s_wa1t[h * FF];
        float4 c0 = wc[0], c1 = wc[1], c2 = wc[2], c3 = wc[3];
        float s = s_ba1[h];
        s += xv[0]*c0.x + xv[1]*c0.y + xv[2]*c0.z + xv[3]*c0.w;
        s += xv[4]*c1.x + xv[5]*c1.y + xv[6]*c1.z + xv[7]*c1.w;
        s += xv[8]*c2.x + xv[9]*c2.y + xv[10]*c2.z + xv[11]*c2.w;
        s += xv[12]*c3.x + xv[13]*c3.y + xv[14]*c3.z + xv[15]*c3.w;
        float a = elu_f(s);
        const float4* wd = (const float4*)&s_wa2[h * FF];
        float4 d0 = wd[0], d1 = wd[1], d2 = wd[2], d3 = wd[3];
        lg[0] += a*d0.x; lg[1] += a*d0.y; lg[2] += a*d0.z; lg[3] += a*d0.w;
        lg[4] += a*d1.x; lg[5] += a*d1.y; lg[6] += a*d1.z; lg[7] += a*d1.w;
        lg[8] += a*d2.x; lg[9] += a*d2.y; lg[10]+= a*d2.z; lg[11]+= a*d2.w;
        lg[12]+= a*d3.x; lg[13]+= a*d3.y; lg[14]+= a*d3.z; lg[15]+= a*d3.w;
    }

    // softmax over F=16
    float m = lg[0];
#pragma unroll
    for (int f = 1; f < 16; ++f) m = fmaxf(m, lg[f]);
    float s = 0.0f;
#pragma unroll
    for (int f = 0; f < 16; ++f) { lg[f] = __expf(lg[f] - m); s += lg[f]; }
    float inv = 1.0f / s;
#pragma unroll
    for (int f = 0; f < 16; ++f) lg[f] *= inv;

    float4* wo = (float4*)(wout + row * FF);
    wo[0] = make_float4(lg[0], lg[1], lg[2], lg[3]);
    wo[1] = make_float4(lg[4], lg[5], lg[6], lg[7]);
    wo[2] = make_float4(lg[8], lg[9], lg[10], lg[11]);
    wo[3] = make_float4(lg[12], lg[13], lg[14], lg[15]);
}

// ---------------------------------------------------------------------------
// Phase 2: out[row,n] = sum_f w[row,f] * ( elu(x[row,f]*W1[f]+b1[f]) @ W2[f] + b2[f] )[n]
// Weight folded into A fragment -> single WMMA f32 accumulator chain over all
// 16 features and K=256.  Block tile: 128 rows x 128 cols, 8 waves
// (wave = 16 rows x 8 N-tiles).  W2 staged in LDS in interleaved K-pair
// layout so each B fragment is one conflict-free ds_load_b64.
// ---------------------------------------------------------------------------
#define NTW   8                 // N-tiles per wave (128 cols)
#define KC    64                // K rows per LDS chunk (32 pairs)
#define PSTR  288               // floats per pair-row: 256 + 32 pad (288%64==32)

__global__ __launch_bounds__(256) void vsn_main_kernel(
    const float* __restrict__ x, const float* __restrict__ W1,
    const float* __restrict__ b1, const float* __restrict__ W2,
    const float* __restrict__ b2, const float* __restrict__ wts,
    float* __restrict__ out)
{
    __shared__ float  s_w2[(KC / 2) * PSTR];  // 32 pairs x 288 floats   (36 KB)
    __shared__ float4 s_wb[128];              // {W1[2k],W1[2k+1],b1[2k],b1[2k+1]} (2 KB)
    __shared__ float  s_wt[128 * FF];         // softmax weights for row block    (8 KB)

    const int tid  = threadIdx.x;
    const int wave = tid >> 5;
    const int lane = tid & 31;
    const int half = lane >> 4;   // A layout: 0 -> K pair (k0,k0+1), 1 -> (k0+2,k0+3)
    const int lr   = lane & 15;
    const int n0   = blockIdx.x * 128;
    const int R0   = blockIdx.y * 128;

    // stage softmax weights for the 128 rows of this block (512 float4)
    {
        const float4* ws = (const float4*)(wts + (size_t)R0 * FF);
        float4* wd = (float4*)s_wt;
        wd[tid]       = ws[tid];
        wd[tid + 256] = ws[tid + 256];
    }
    // x values for my wave's 16 rows (lane lr owns row R0+wave*16+lr)
    const int myrow = R0 + wave * 16 + lr;
    const float4* xs = (const float4*)(x + (size_t)myrow * FF);
    float4 x0 = xs[0], x1 = xs[1], x2 = xs[2], x3 = xs[3];
    float xv[16] = {x0.x, x0.y, x0.z, x0.w, x1.x, x1.y, x1.z, x1.w,
                    x2.x, x2.y, x2.z, x2.w, x3.x, x3.y, x3.z, x3.w};

    v8f acc[NTW];
#pragma unroll
    for (int t = 0; t < NTW; ++t) acc[t] = (v8f)0.0f;

    __syncthreads();

    for (int f = 0; f < FF; ++f) {
        const float wa = s_wt[(wave * 16 + lr) * FF + f];  // row weight, folded into A
        const float xf = xv[f];

        for (int kc = 0; kc < HH / KC; ++kc) {   // 4 K-chunks of 64 rows
            __syncthreads();
            // cooperative load W2[f][kc*64 .. +63][n0 .. n0+127] into LDS,
            // interleaving each K-pair: s_w2[p*PSTR + 2n + {0,1}] = W2[2p / 2p+1][n]
            {
                const float* w2base =
                    W2 + ((size_t)f * HH + kc * KC) * HH + n0;
#pragma unroll
                for (int i = 0; i < 4; ++i) {
                    int idx = tid + i * 256;      // 0..1023 = 32 pairs x 32 quads
                    int p = idx >> 5;             // pair index 0..31
                    int q = idx & 31;             // 4-col group 0..31
                    float4 r0 = *(const float4*)(w2base + (size_t)(2 * p) * HH + 4 * q);
                    float4 r1 = *(const float4*)(w2base + (size_t)(2 * p + 1) * HH + 4 * q);
                    float* d = &s_w2[p * PSTR + 8 * q];
                    *(float4*)(d)     = make_float4(r0.x, r1.x, r0.y, r1.y);
                    *(float4*)(d + 4) = make_float4(r0.z, r1.z, r0.w, r1.w);
                }
                if (kc == 0 && tid < 128) {       // pack W1/b1 k-pairs for this f
                    float2 w  = *(const float2*)(W1 + f * HH + tid * 2);
                    float2 bb = *(const float2*)(b1 + f * HH + tid * 2);
                    s_wb[tid] = make_float4(w.x, w.y, bb.x, bb.y);
                }
            }
            __syncthreads();

#pragma unroll 4
            for (int ks = 0; ks < KC / 4; ++ks) {   // 16 K-steps of 4
                // A fragment: a = w * elu(x_f * W1 + b1) for my two K values
                float4 p4 = s_wb[kc * (KC / 2) + 2 * ks + half];
                v2f afrag;
                afrag.x = wa * elu_f(xf * p4.x + p4.z);
                afrag.y = wa * elu_f(xf * p4.y + p4.w);
                // B fragments: one ds_load_b64 per N-tile (interleaved pair)
                const int prow = 2 * ks + half;     // local pair index
#pragma unroll
                for (int t = 0; t < NTW; ++t) {
                    v2f bfrag = *(const v2f*)&s_w2[prow * PSTR + 2 * (t * 16 + lr)];
                    acc[t] = __builtin_amdgcn_wmma_f32_16x16x4_f32(
                        false, afrag, false, bfrag, (short)0, acc[t],
                        false, false);
                }
            }
        }

        // bias contribution: acc[t][r] += w[row(r)] * b2[f][n(t)]
        float wr[8];
#pragma unroll
        for (int r = 0; r < 8; ++r)
            wr[r] = s_wt[(wave * 16 + 8 * half + r) * FF + f];
#pragma unroll
        for (int t = 0; t < NTW; ++t) {
            float b2v = b2[f * HH + n0 + t * 16 + lr];
#pragma unroll
            for (int r = 0; r < 8; ++r) acc[t][r] += wr[r] * b2v;
        }
    }

    // store: C/D layout -> VGPR r, lane l: row = 8*(l/16)+r, col = n0+16t+(l%16)
#pragma unroll
    for (int t = 0; t < NTW; ++t) {
#pragma unroll
        for (int r = 0; r < 8; ++r) {
            size_t row = (size_t)(R0 + wave * 16 + 8 * half + r);
            out[row * HH + n0 + t * 16 + lr] = acc[t][r];
        }
    }
}

// ---------------------------------------------------------------------------
extern "C" void kernel_launch(void* const* d_in, const int* in_sizes, int n_in,
                              void* d_out, int out_size, void* d_ws, size_t ws_size,
                              hipStream_t stream) {
    const float* x   = (const float*)d_in[0];
    const float* W1  = (const float*)d_in[1];
    const float* b1  = (const float*)d_in[2];
    const float* W2  = (const float*)d_in[3];
    const float* b2  = (const float*)d_in[4];
    const float* Wa1 = (const float*)d_in[5];
    const float* ba1 = (const float*)d_in[6];
    const float* Wa2 = (const float*)d_in[7];
    const float* ba2 = (const float*)d_in[8];

    float* out  = (float*)d_out;                    // [BS, 256] main output
    float* wout = out + (size_t)BS * HH;            // [BS, 16] weights output

    // Phase 1: attention weights (also second output)
    vsn_attn_kernel<<<dim3(BS / 256), dim3(256), 0, stream>>>(
        x, Wa1, ba1, Wa2, ba2, wout);

    // Phase 2: weighted per-feature FFN via fp32 WMMA
    vsn_main_kernel<<<dim3(HH / 128, BS / 128), dim3(256), 0, stream>>>(
        x, W1, b1, W2, b2, wout, out);
}